// PairTabModel_63599875719181
// MI455X (gfx1250) — compile-verified
//
#include <hip/hip_runtime.h>
#include <stdint.h>

// Problem constants (from reference)
#define NFRAMES 8
#define NLOC    8192
#define NALL    12288
#define NNEI    256
#define NTYPES  4
#define NSPLINE 2000
// RMIN = 0.0, HH = 0.01  ->  uu = rr * 100.0f

// Launch shape
#define THREADS          1024            // 32 waves (wave32)
#define CHUNKS           32              // blocks per frame
#define ATOMS_PER_BLOCK  (NLOC / CHUNKS) // 256
#define WAVES            (THREADS / 32)  // 32
#define ATOMS_PER_WAVE   (ATOMS_PER_BLOCK / WAVES) // 8

// LDS: one float4 record per extended atom: {x, y, z, bitcast(type)}
#define SMEM_BYTES (NALL * 16)           // 196608 bytes (<= 320KB WGP LDS)

// Raw 32-bit LDS byte offset of a shared-memory pointer.
typedef __attribute__((address_space(3))) const void lds_cv_t;
__device__ __forceinline__ unsigned lds_off(const void* p) {
    return (unsigned)(unsigned long long)(lds_cv_t*)p;
}

extern "C" __global__ __launch_bounds__(THREADS)
void pairtab_kernel(const float* __restrict__ coord,   // (NFRAMES, NALL, 3)
                    const int*   __restrict__ atype,   // (NFRAMES, NALL)
                    const int*   __restrict__ nlist,   // (NFRAMES, NLOC, NNEI)
                    const float* __restrict__ tab,     // (NTYPES, NTYPES, NSPLINE, 4)
                    float*       __restrict__ out)     // (NFRAMES, NLOC)
{
    extern __shared__ float smem[];
    const float4* s_pack = (const float4*)smem;        // NALL records of 16B

    const int frame = blockIdx.y;
    const int chunk = blockIdx.x;
    const int tid   = threadIdx.x;
    const int lane  = tid & 31;
    const int wave  = tid >> 5;

    // ---- Async-build interleaved {x,y,z,type} records in LDS ----
    // IOFFSET adds to BOTH the LDS and global addresses, so x/y/z share one
    // (lds_base, gvoff) pair with offset:0/4/8; the type slot gets its own.
    {
        const unsigned long long gc =
            (unsigned long long)(const void*)(coord + (size_t)frame * (NALL * 3));
        const unsigned long long gt =
            (unsigned long long)(const void*)(atype + (size_t)frame * NALL);
        const unsigned lbase = lds_off(smem);
        #pragma unroll
        for (int it = 0; it < NALL / THREADS; ++it) {  // 12 atoms per thread
            const unsigned a   = (unsigned)(tid + it * THREADS);
            const unsigned l   = lbase + a * 16u;      // LDS record base
            const unsigned gcb = a * 12u;              // global coord byte off
            const unsigned gtb = a * 4u;               // global type byte off
            const unsigned l12 = l + 12u;
            asm volatile("global_load_async_to_lds_b32 %0, %1, %2 offset:0"
                         :: "v"(l), "v"(gcb), "s"(gc) : "memory");
            asm volatile("global_load_async_to_lds_b32 %0, %1, %2 offset:4"
                         :: "v"(l), "v"(gcb), "s"(gc) : "memory");
            asm volatile("global_load_async_to_lds_b32 %0, %1, %2 offset:8"
                         :: "v"(l), "v"(gcb), "s"(gc) : "memory");
            asm volatile("global_load_async_to_lds_b32 %0, %1, %2 offset:0"
                         :: "v"(l12), "v"(gtb), "s"(gt) : "memory");
        }
        asm volatile("s_wait_asynccnt 0" ::: "memory");
    }
    __syncthreads();

    const float4* tab4 = (const float4*)tab;

    // ---- One wave per atom; each lane owns 8 neighbors (2 x int4) ----
    #pragma unroll 2
    for (int t = 0; t < ATOMS_PER_WAVE; ++t) {
        const int iloc = chunk * ATOMS_PER_BLOCK + t * WAVES + wave;

        const float4 pi = s_pack[iloc];                // ds_load_b128
        const int    ity = __float_as_int(pi.w);
        const int    ityBase = ity * (NTYPES * NSPLINE);

        const int* nrow = nlist + ((size_t)frame * NLOC + iloc) * NNEI;
        if (t + 1 < ATOMS_PER_WAVE) {
            // prefetch next atom's nlist row (1024B = 8 cachelines)
            __builtin_prefetch(nrow + (size_t)WAVES * NNEI + lane * 32, 0, 0);
        }
        const int4* nrow4 = (const int4*)nrow;

        float acc = 0.0f;
        #pragma unroll
        for (int h = 0; h < 2; ++h) {
            const int4 nj = nrow4[h * 32 + lane];      // coalesced b128
            const int js[4] = {nj.x, nj.y, nj.z, nj.w};
            #pragma unroll
            for (int q = 0; q < 4; ++q) {
                const int j  = js[q];
                const int jj = (j < 0) ? 0 : j;        // v_max_i32

                const float4 pj = s_pack[jj];          // single ds_load_b128
                const int    jt = __float_as_int(pj.w);

                const float dx = pi.x - pj.x;
                const float dy = pi.y - pj.y;
                const float dz = pi.z - pj.z;
                const float rr = sqrtf(dx * dx + dy * dy + dz * dz);

                const float uu  = rr * 100.0f;         // (rr - RMIN)/HH
                const int   idx = (int)uu;             // uu >= 0 -> trunc == floor
                const float u   = uu - (float)idx;
                const int   ci  = idx < (NSPLINE - 1) ? idx : (NSPLINE - 1);

                const float4 c = tab4[ityBase + jt * NSPLINE + ci]; // L2 b128 gather
                const float  e = ((c.x * u + c.y) * u + c.z) * u + c.w;

                const bool keep = (j >= 0) & (idx < NSPLINE);
                acc += keep ? e : 0.0f;
            }
        }

        // wave32 reduction over neighbors
        acc += __shfl_xor(acc, 16, 32);
        acc += __shfl_xor(acc,  8, 32);
        acc += __shfl_xor(acc,  4, 32);
        acc += __shfl_xor(acc,  2, 32);
        acc += __shfl_xor(acc,  1, 32);

        if (lane == 0) out[(size_t)frame * NLOC + iloc] = 0.5f * acc;
    }
}

extern "C" void kernel_launch(void* const* d_in, const int* in_sizes, int n_in,
                              void* d_out, int out_size, void* d_ws, size_t ws_size,
                              hipStream_t stream) {
    const float* coord = (const float*)d_in[0]; // extended_coord
    const int*   atyp  = (const int*)  d_in[1]; // extended_atype
    const int*   nlst  = (const int*)  d_in[2]; // nlist
    const float* tab   = (const float*)d_in[3]; // tab_data
    float*       out   = (float*)d_out;

    // Allow >48KB dynamic LDS (gfx1250 WGP has 320KB)
    (void)hipFuncSetAttribute((const void*)pairtab_kernel,
                              hipFuncAttributeMaxDynamicSharedMemorySize,
                              SMEM_BYTES);

    dim3 grid(CHUNKS, NFRAMES);
    pairtab_kernel<<<grid, THREADS, SMEM_BYTES, stream>>>(coord, atyp, nlst, tab, out);
}